// LDDTMetrics_84817014161962
// MI455X (gfx1250) — compile-verified
//
#include <hip/hip_runtime.h>
#include <stdint.h>

#define N_SAMPLE 5
#define N_ATOM   20000
#define N_PAIRS  4000000
#define N_GROUPS (N_PAIRS / 4)

#define MAIN_BLOCKS  256
#define MAIN_THREADS 1024

// true_coordinate staged in LDS padded to float4/atom: 20000*16 = 320000 B
// (fits the 320 KB WGP LDS; one workgroup per WGP, 32 waves = 8 per SIMD32)
#define TRUE_LDS_B   (N_ATOM * 16)

// ---------------------------------------------------------------------------
// gfx1250 async global->LDS copy (async data-mover path), with fallback.
// Params (confirmed via hipcc diagnostics): (src addrspace(1)*,
// dst addrspace(3)*, imm offset, imm cpol), element-typed pointers.
// ---------------------------------------------------------------------------
typedef __attribute__((address_space(1))) int* gas_i32_ptr;   // global AS
typedef __attribute__((address_space(3))) int* las_i32_ptr;   // LDS AS

__device__ __forceinline__ void async_copy_b32(const float* gsrc, float* ldst) {
#if __has_builtin(__builtin_amdgcn_global_load_async_to_lds_b32)
    __builtin_amdgcn_global_load_async_to_lds_b32(
        (gas_i32_ptr)gsrc, (las_i32_ptr)ldst, 0, 0);
#else
    *ldst = *gsrc;   // synchronous fallback
#endif
}

__device__ __forceinline__ void wait_async_copies() {
#if __has_builtin(__builtin_amdgcn_s_wait_asynccnt)
    __builtin_amdgcn_s_wait_asynccnt(0);
#else
    asm volatile("s_wait_asynccnt 0" ::: "memory");
#endif
}

__device__ __forceinline__ float dist3(float dx, float dy, float dz) {
    return sqrtf(fmaf(dx, dx, fmaf(dy, dy, dz * dz)));
}

__device__ __forceinline__ void cnt_add(unsigned& acc, float dx, float dy,
                                        float dz, float td) {
    const float pd  = dist3(dx, dy, dz);
    const float err = fabsf(pd - td);
    acc += (unsigned)(err < 0.5f) + (unsigned)(err < 1.0f) +
           (unsigned)(err < 2.0f) + (unsigned)(err < 4.0f);
}

// ---------------------------------------------------------------------------
// Main gather + distance + threshold-count kernel.
// pred_pk: one 64-byte record per atom = 5 samples x packed xyz (60 B) + pad.
// Each endpoint gather touches exactly ONE cacheline (4x global_load_b128).
// ---------------------------------------------------------------------------
__global__ void __launch_bounds__(MAIN_THREADS, 1)
lddt_main_kernel(const float4* __restrict__ pred_pk,    // [A][4] float4 (64 B/atom)
                 const float*  __restrict__ true_c,     // [A][3]
                 const int*    __restrict__ l_index,
                 const int*    __restrict__ m_index,
                 unsigned*     __restrict__ g_cnt) {
    extern __shared__ float4 s_true4[];                 // 320000 B dynamic LDS
    float* s_true_f = (float*)s_true4;
    __shared__ unsigned s_cnt[N_SAMPLE];

    const int tid = threadIdx.x;
    if (tid < N_SAMPLE) s_cnt[tid] = 0u;

    // --- stage true coords into padded-LDS via async global->LDS b32 ---
    for (int a = tid; a < N_ATOM; a += MAIN_THREADS) {
        const float* src = true_c + a * 3;
        float* dst = s_true_f + a * 4;
        async_copy_b32(src + 0, dst + 0);
        async_copy_b32(src + 1, dst + 1);
        async_copy_b32(src + 2, dst + 2);
    }
    wait_async_copies();
    __syncthreads();

    unsigned acc[N_SAMPLE];
#pragma unroll
    for (int s = 0; s < N_SAMPLE; ++s) acc[s] = 0u;

    const int4* l4 = (const int4*)l_index;
    const int4* m4 = (const int4*)m_index;
    const int gsize = gridDim.x * MAIN_THREADS;
    int g = blockIdx.x * MAIN_THREADS + tid;

    for (; g < N_GROUPS; g += gsize) {
        // speculative prefetch of the next index chunk (global_prefetch_b8)
        __builtin_prefetch(l4 + g + gsize, 0, 0);
        __builtin_prefetch(m4 + g + gsize, 0, 0);

        const int4 lv = l4[g];
        const int4 mv = m4[g];
        const int lis[4] = {lv.x, lv.y, lv.z, lv.w};
        const int mis[4] = {mv.x, mv.y, mv.z, mv.w};

#pragma unroll
        for (int k = 0; k < 4; ++k) {
            const int li = lis[k];
            const int mi = mis[k];

            // true distance: 2x ds_load_b128, reused across all 5 samples
            const float4 tl = s_true4[li];
            const float4 tm = s_true4[mi];
            const float td = dist3(tl.x - tm.x, tl.y - tm.y, tl.z - tm.z);

            // one 64 B record per endpoint: 4x global_load_b128, 1 cacheline
            const float4* Pl = pred_pk + (size_t)li * 4;
            const float4* Pm = pred_pk + (size_t)mi * 4;
            const float4 L0 = Pl[0], L1 = Pl[1], L2 = Pl[2], L3 = Pl[3];
            const float4 M0 = Pm[0], M1 = Pm[1], M2 = Pm[2], M3 = Pm[3];

            // record layout: [s0.xyz s1.xyz s2.xyz s3.xyz s4.xyz pad]
            cnt_add(acc[0], L0.x - M0.x, L0.y - M0.y, L0.z - M0.z, td);
            cnt_add(acc[1], L0.w - M0.w, L1.x - M1.x, L1.y - M1.y, td);
            cnt_add(acc[2], L1.z - M1.z, L1.w - M1.w, L2.x - M2.x, td);
            cnt_add(acc[3], L2.y - M2.y, L2.z - M2.z, L2.w - M2.w, td);
            cnt_add(acc[4], L3.x - M3.x, L3.y - M3.y, L3.z - M3.z, td);
        }
    }

    // --- wave32 reduction (warpSize == 32 on gfx1250) ---
#pragma unroll
    for (int s = 0; s < N_SAMPLE; ++s) {
        unsigned v = acc[s];
#pragma unroll
        for (int off = 16; off > 0; off >>= 1)
            v += (unsigned)__shfl_xor((int)v, off, 32);
        if ((tid & 31) == 0) atomicAdd(&s_cnt[s], v);
    }
    __syncthreads();
    if (tid < N_SAMPLE) atomicAdd(&g_cnt[tid], s_cnt[tid]);
}

// ---------------------------------------------------------------------------
// Zero the 5 global counters in workspace
// ---------------------------------------------------------------------------
__global__ void lddt_init_kernel(unsigned* cnt) {
    if (threadIdx.x < N_SAMPLE) cnt[threadIdx.x] = 0u;
}

// ---------------------------------------------------------------------------
// Repack pred [S][A][3] f32 -> per-atom 64 B record:
//   [s0.xyz s1.xyz s2.xyz s3.xyz s4.xyz pad] (15 floats + 1 pad)
// ---------------------------------------------------------------------------
__global__ void lddt_repack_kernel(const float* __restrict__ pred,
                                   float4* __restrict__ pred_pk) {
    const int a = blockIdx.x * blockDim.x + threadIdx.x;
    if (a >= N_ATOM) return;
    float r[16];
#pragma unroll
    for (int s = 0; s < N_SAMPLE; ++s) {
        const float* src = pred + ((size_t)s * N_ATOM + a) * 3;
        r[s * 3 + 0] = src[0];
        r[s * 3 + 1] = src[1];
        r[s * 3 + 2] = src[2];
    }
    r[15] = 0.0f;
    float4* dst = pred_pk + (size_t)a * 4;
    dst[0] = make_float4(r[0],  r[1],  r[2],  r[3]);
    dst[1] = make_float4(r[4],  r[5],  r[6],  r[7]);
    dst[2] = make_float4(r[8],  r[9],  r[10], r[11]);
    dst[3] = make_float4(r[12], r[13], r[14], r[15]);
}

// ---------------------------------------------------------------------------
// Finalize: lddt[s] = count[s] / (4 * N_PAIRS)
// ---------------------------------------------------------------------------
__global__ void lddt_finalize_kernel(const unsigned* __restrict__ cnt,
                                     float* __restrict__ out) {
    if (threadIdx.x < N_SAMPLE)
        out[threadIdx.x] =
            (float)((double)cnt[threadIdx.x] / (4.0 * (double)N_PAIRS));
}

// ---------------------------------------------------------------------------
extern "C" void kernel_launch(void* const* d_in, const int* in_sizes, int n_in,
                              void* d_out, int out_size, void* d_ws, size_t ws_size,
                              hipStream_t stream) {
    const float* pred   = (const float*)d_in[0];  // [5][20000][3]
    const float* true_c = (const float*)d_in[1];  // [20000][3]
    const int*   l_idx  = (const int*)d_in[2];    // [4M]
    const int*   m_idx  = (const int*)d_in[3];    // [4M]

    // workspace: [0,20) counters (5 x u32), pad to 256, then packed pred records
    unsigned* cnt     = (unsigned*)d_ws;
    float4*   pred_pk = (float4*)((char*)d_ws + 256);  // 20000 * 64 B = 1.28 MB

    lddt_init_kernel<<<1, 32, 0, stream>>>(cnt);

    lddt_repack_kernel<<<(N_ATOM + 255) / 256, 256, 0, stream>>>(pred, pred_pk);

    lddt_main_kernel<<<MAIN_BLOCKS, MAIN_THREADS, TRUE_LDS_B, stream>>>(
        pred_pk, true_c, l_idx, m_idx, cnt);

    lddt_finalize_kernel<<<1, 32, 0, stream>>>(cnt, (float*)d_out);
}